// AttentionPooling_vectorized_old_46059229282479
// MI455X (gfx1250) — compile-verified
//
#include <hip/hip_runtime.h>
#include <math.h>

#define B_   4
#define S_   512
#define NS_  8192
#define MAXW 16
#define H_   256
#define NH_  4
#define DH_  64
#define FF_  1024
#define NEGV (-1000000000.0f)
#define LDA  257   // padded LDS row stride (breaks bank conflicts on column reads)

typedef __attribute__((ext_vector_type(2))) float v2f;
typedef __attribute__((ext_vector_type(8))) float v8f;

// ---------------------------------------------------------------------------
// Two 16x16 fp32 WMMA tiles (n0 and n0+16) sharing one A-operand stream:
//   D0 += A(16xK) * B[:, n0:n0+16],  D1 += A * B[:, n0+16:n0+32]
// B element [k][n] lives at W[k*sK + n*sN].  Two independent accumulator
// chains interleave in the XDL pipe; A is loaded from LDS once per k-step.
// ---------------------------------------------------------------------------
__device__ inline void gemm_tile2(const float* __restrict__ A, int lda, int K,
                                  const float* __restrict__ W, int sK, int sN,
                                  int n0, v8f& acc0, v8f& acc1) {
  const int lane = threadIdx.x & 31;
  const int half = lane >> 4;
  const int lr   = lane & 15;
  const float* Ar  = A + lr * lda + 2 * half;
  const float* Wn0 = W + (size_t)(n0 + lr) * sN + (size_t)(2 * half) * sK;
  const float* Wn1 = Wn0 + (size_t)16 * sN;
#pragma unroll 4
  for (int k = 0; k < K; k += 4) {
    v2f a;  a.x  = Ar[k];                 a.y  = Ar[k + 1];
    v2f b0; b0.x = Wn0[(size_t)k * sK];   b0.y = Wn0[(size_t)(k + 1) * sK];
    v2f b1; b1.x = Wn1[(size_t)k * sK];   b1.y = Wn1[(size_t)(k + 1) * sK];
    acc0 = __builtin_amdgcn_wmma_f32_16x16x4_f32(false, a, false, b0,
                                                 (short)0, acc0, false, false);
    acc1 = __builtin_amdgcn_wmma_f32_16x16x4_f32(false, a, false, b1,
                                                 (short)0, acc1, false, false);
  }
}

__device__ inline v8f zero_v8f() {
  v8f z = {0.f, 0.f, 0.f, 0.f, 0.f, 0.f, 0.f, 0.f};
  return z;
}

// ---------------------------------------------------------------------------
// q = (dummy_query @ w_q^T + b_q) / sqrt(DH)
// ---------------------------------------------------------------------------
__global__ void qvec_kernel(const float* __restrict__ dq,
                            const float* __restrict__ wq,
                            const float* __restrict__ bqkv,
                            float* __restrict__ qs) {
  int t = threadIdx.x;
  float s = 0.f;
  const float* wr = wq + (size_t)t * H_;
  for (int j = 0; j < H_; ++j) s += wr[j] * dq[j];
  qs[t] = (s + bqkv[t]) * 0.125f;  // 1/sqrt(64)
}

// ---------------------------------------------------------------------------
// wo_t[k][n] = wo[n][k] : makes GEMM1's B loads lane-coalesced.
// 16x16 tiles through LDS; grid = 256 blocks of 256 threads.
// ---------------------------------------------------------------------------
__global__ void transpose_wo_kernel(const float* __restrict__ wo,
                                    float* __restrict__ wot) {
  __shared__ float tile[16][17];
  int bx = blockIdx.x & 15, by = blockIdx.x >> 4;
  int tx = threadIdx.x & 15, ty = threadIdx.x >> 4;
  tile[ty][tx] = wo[(size_t)(by * 16 + ty) * H_ + bx * 16 + tx];
  __syncthreads();
  wot[(size_t)(bx * 16 + ty) * H_ + by * 16 + tx] = tile[tx][ty];
}

// ---------------------------------------------------------------------------
// Per token row: tr = tok + PE;  v_tok = tr @ w_v^T + b_v;
// logit_tok[h] = (q/8) . (tr @ w_k^T + b_k) per 64-dim head chunk.
// ---------------------------------------------------------------------------
__global__ __launch_bounds__(256) void prep_kernel(
    const float* __restrict__ tok, const float* __restrict__ wk,
    const float* __restrict__ wv,  const float* __restrict__ bqkv,
    const float* __restrict__ qs,
    float* __restrict__ vtok, float* __restrict__ ltok) {
  int blk = blockIdx.x;
  int b = blk / S_, s = blk % S_;
  int t = threadIdx.x;
  __shared__ float tr[H_];
  __shared__ float kq[H_];

  // sinusoidal positional encoding, matching numpy reference
  float freq = expf((float)(t & ~1) * (-9.210340371976184f / 256.0f));
  float ang  = (float)s * freq;
  float pe   = (t & 1) ? cosf(ang) : sinf(ang);
  tr[t] = tok[(size_t)(b * S_ + s) * H_ + t] + pe;
  __syncthreads();

  float kk = 0.f, vk = 0.f;
  const float* wkr = wk + (size_t)t * H_;
  const float* wvr = wv + (size_t)t * H_;
  for (int j = 0; j < H_; ++j) {
    float x = tr[j];
    kk += wkr[j] * x;
    vk += wvr[j] * x;
  }
  kk += bqkv[H_ + t];       // b_qkv[1]
  vk += bqkv[2 * H_ + t];   // b_qkv[2]
  vtok[(size_t)(b * S_ + s) * H_ + t] = vk;
  kq[t] = kk * qs[t];
  __syncthreads();
  if (t < NH_) {
    float acc = 0.f;
    for (int d = 0; d < DH_; ++d) acc += kq[t * DH_ + d];
    ltok[(size_t)(b * S_ + s) * NH_ + t] = acc;
  }
}

// ---------------------------------------------------------------------------
// Fused: span gather + softmax + ctx  ->  GEMM1(+b_o+q) -> LN -> FFN (WMMA,
// 4 phases of 256 hidden cols, GEMM3 accumulators held in VGPRs) -> LN -> mask
// One block = 16 spans (rows), 256 threads = 8 waves; each wave owns a
// 32-column strip (two 16x16 WMMA tiles with interleaved accumulators).
// ---------------------------------------------------------------------------
__global__ __launch_bounds__(256) void fused_kernel(
    const int* __restrict__ span_ids, const unsigned char* __restrict__ span_masks,
    const float* __restrict__ ltok, const float* __restrict__ vtok,
    const float* __restrict__ dq,
    const float* __restrict__ wot, const float* __restrict__ bo,
    const float* __restrict__ lng, const float* __restrict__ lnb,
    const float* __restrict__ w1, const float* __restrict__ b1,
    const float* __restrict__ w2, const float* __restrict__ b2,
    float* __restrict__ out) {
  __shared__ float bufA[16 * LDA];  // ctx, then y1 (post-LN residual)
  __shared__ float bufB[16 * LDA];  // attn-out, then ffn-out
  __shared__ float bufH[16 * LDA];  // ffn hidden chunk (256 cols per phase)
  __shared__ float red[256], red2[256];
  __shared__ float mean_s[16], rstd_s[16];
  __shared__ int   s0a[16], lena[16], mska[16];
  __shared__ float lw[16 * 64];     // logits then attn weights

  const int t    = threadIdx.x;
  const int r0   = blockIdx.x * 16;          // first global span row
  const int b    = r0 / NS_;                 // NS_ % 16 == 0 -> uniform
  const int wid  = t >> 5;
  const int lane = t & 31;
  const int half = lane >> 4, lr = lane & 15;

  if (t < 16) {
    int m  = r0 + t;
    int s0 = span_ids[2 * m];
    int s1 = span_ids[2 * m + 1];
    int mk = span_masks[m] ? 1 : 0;
    s0a[t]  = s0;
    lena[t] = mk ? (s1 - s0) : 0;
    mska[t] = mk;
  }
  __syncthreads();

  // ---- gather per-head logits for 16 spans x 4 heads x 16 positions ----
  for (int rep = 0; rep < 4; ++rep) {
    int task = rep * 256 + t;
    int sp = task >> 6;
    int hw = task & 63;
    int h = hw >> 4, w = hw & 15;
    int pos = s0a[sp] + w;
    pos = min(max(pos, 0), S_ - 1);
    float lg = (w < lena[sp]) ? ltok[(size_t)(b * S_ + pos) * NH_ + h] : NEGV;
    lw[sp * 64 + hw] = lg;
  }
  __syncthreads();

  // ---- softmax over window, one thread per (span, head) ----
  if (t < 64) {
    int sp = t >> 2, h = t & 3;
    float* p = lw + sp * 64 + h * 16;
    float mx = p[0];
    for (int w = 1; w < 16; ++w) mx = fmaxf(mx, p[w]);
    float e[16], sum = 0.f;
    for (int w = 0; w < 16; ++w) { e[w] = expf(p[w] - mx); sum += e[w]; }
    float inv = 1.f / sum;
    for (int w = 0; w < 16; ++w) p[w] = e[w] * inv;
  }
  __syncthreads();

  // ---- ctx tile (16x256) straight into LDS; coalesced over channels ----
  for (int sp = 0; sp < 16; ++sp) {
    int c = t;
    int h = c >> 6;
    const float* at = lw + sp * 64 + h * 16;
    int s0 = s0a[sp];
    float acc = 0.f;
#pragma unroll
    for (int w = 0; w < 16; ++w) {
      int pos = min(max(s0 + w, 0), S_ - 1);
      acc += at[w] * vtok[(size_t)(b * S_ + pos) * H_ + c];
    }
    bufA[sp * LDA + c] = acc;
  }
  __syncthreads();

  // ---- GEMM1: out = ctx @ w_o^T + b_o + dummy_query -> bufB ----
  {
    v8f acc0 = zero_v8f(), acc1 = zero_v8f();
    gemm_tile2(bufA, LDA, H_, wot, /*sK=*/H_, /*sN=*/1, wid * 32, acc0, acc1);
#pragma unroll
    for (int tile = 0; tile < 2; ++tile) {
      int n = wid * 32 + tile * 16 + lr;
      float add = bo[n] + dq[n];
      const v8f& acc = tile ? acc1 : acc0;
#pragma unroll
      for (int j = 0; j < 8; ++j)
        bufB[(j + 8 * half) * LDA + n] = acc[j] + add;
    }
  }
  __syncthreads();

  // ---- LayerNorm 1: bufB -> bufA (y1) ----
  {
    int row = t >> 4, part = t & 15;
    float s = 0.f, s2 = 0.f;
    const float* p = bufB + row * LDA + part * 16;
    for (int i = 0; i < 16; ++i) { float x = p[i]; s += x; s2 += x * x; }
    red[t] = s; red2[t] = s2;
  }
  __syncthreads();
  if (t < 16) {
    float s = 0.f, s2 = 0.f;
    for (int i = 0; i < 16; ++i) { s += red[t * 16 + i]; s2 += red2[t * 16 + i]; }
    float mean = s * (1.f / H_);
    float var  = s2 * (1.f / H_) - mean * mean;
    mean_s[t] = mean; rstd_s[t] = rsqrtf(var + 1e-5f);
  }
  __syncthreads();
  for (int i = 0; i < 16; ++i) {
    int idx = t + i * 256, row = idx >> 8, col = idx & 255;
    bufA[row * LDA + col] =
        (bufB[row * LDA + col] - mean_s[row]) * rstd_s[row] * lng[col] + lnb[col];
  }
  __syncthreads();

  // ---- FFN in 4 phases of 256 hidden cols; GEMM3 acc stays in VGPRs ----
  v8f acc3_0 = zero_v8f();
  v8f acc3_1 = zero_v8f();
  for (int ph = 0; ph < 4; ++ph) {
    // GEMM2 chunk: h[:, ph*256 : ph*256+256] = relu(y1 @ w1 + b1) -> bufH
    {
      int nloc = wid * 32;
      int ng   = ph * 256 + nloc;
      v8f h0 = zero_v8f(), h1 = zero_v8f();
      gemm_tile2(bufA, LDA, H_, w1, /*sK=*/FF_, /*sN=*/1, ng, h0, h1);
#pragma unroll
      for (int tile = 0; tile < 2; ++tile) {
        int nl = nloc + tile * 16 + lr;
        float bb = b1[ph * 256 + nl];
        const v8f& acc = tile ? h1 : h0;
#pragma unroll
        for (int j = 0; j < 8; ++j)
          bufH[(j + 8 * half) * LDA + nl] = fmaxf(acc[j] + bb, 0.f);
      }
    }
    __syncthreads();
    // GEMM3 partial: accumulate h_chunk @ w2[ph*256:ph*256+256, :]
    gemm_tile2(bufH, LDA, 256, w2 + (size_t)(ph * 256) * H_,
               /*sK=*/H_, /*sN=*/1, wid * 32, acc3_0, acc3_1);
    __syncthreads();  // before next phase overwrites bufH
  }

  // ---- epilogue GEMM3: + b2 + residual y1 -> bufB ----
#pragma unroll
  for (int tile = 0; tile < 2; ++tile) {
    int n = wid * 32 + tile * 16 + lr;
    float bb = b2[n];
    const v8f& acc = tile ? acc3_1 : acc3_0;
#pragma unroll
    for (int j = 0; j < 8; ++j) {
      int row = j + 8 * half;
      bufB[row * LDA + n] = acc[j] + bb + bufA[row * LDA + n];
    }
  }
  __syncthreads();

  // ---- LayerNorm 2 + span mask -> out ----
  {
    int row = t >> 4, part = t & 15;
    float s = 0.f, s2 = 0.f;
    const float* p = bufB + row * LDA + part * 16;
    for (int i = 0; i < 16; ++i) { float x = p[i]; s += x; s2 += x * x; }
    red[t] = s; red2[t] = s2;
  }
  __syncthreads();
  if (t < 16) {
    float s = 0.f, s2 = 0.f;
    for (int i = 0; i < 16; ++i) { s += red[t * 16 + i]; s2 += red2[t * 16 + i]; }
    float mean = s * (1.f / H_);
    float var  = s2 * (1.f / H_) - mean * mean;
    mean_s[t] = mean; rstd_s[t] = rsqrtf(var + 1e-5f);
  }
  __syncthreads();
  for (int i = 0; i < 16; ++i) {
    int idx = t + i * 256, row = idx >> 8, col = idx & 255;
    float y =
        (bufB[row * LDA + col] - mean_s[row]) * rstd_s[row] * lng[col] + lnb[col];
    if (!mska[row]) y = 0.f;
    out[(size_t)(r0 + row) * H_ + col] = y;
  }
}

// ---------------------------------------------------------------------------
extern "C" void kernel_launch(void* const* d_in, const int* in_sizes, int n_in,
                              void* d_out, int out_size, void* d_ws, size_t ws_size,
                              hipStream_t stream) {
  const float*         tok        = (const float*)d_in[0];
  const int*           span_ids   = (const int*)d_in[1];
  const unsigned char* span_masks = (const unsigned char*)d_in[2];
  /* d_in[3] = pooling (unused) */
  const float* dq   = (const float*)d_in[4];
  const float* wq   = (const float*)d_in[5];
  const float* wk   = (const float*)d_in[6];
  const float* wv   = (const float*)d_in[7];
  const float* bqkv = (const float*)d_in[8];
  const float* wo   = (const float*)d_in[9];
  const float* bo   = (const float*)d_in[10];
  const float* lng  = (const float*)d_in[11];
  const float* lnb  = (const float*)d_in[12];
  const float* w1   = (const float*)d_in[13];
  const float* b1   = (const float*)d_in[14];
  const float* w2   = (const float*)d_in[15];
  const float* b2   = (const float*)d_in[16];
  float* out = (float*)d_out;

  float* ws   = (float*)d_ws;
  float* qs   = ws;                                         // 256
  float* ltok = ws + 256;                                   // B*S*NH = 8192
  float* vtok = ws + 256 + (size_t)B_ * S_ * NH_;           // B*S*H  = 524288
  float* wot  = vtok + (size_t)B_ * S_ * H_;                // H*H    = 65536

  qvec_kernel<<<1, 256, 0, stream>>>(dq, wq, bqkv, qs);
  transpose_wo_kernel<<<256, 256, 0, stream>>>(wo, wot);
  prep_kernel<<<B_ * S_, 256, 0, stream>>>(tok, wk, wv, bqkv, qs, vtok, ltok);
  fused_kernel<<<(B_ * NS_) / 16, 256, 0, stream>>>(
      span_ids, span_masks, ltok, vtok, dq, wot, bo, lng, lnb, w1, b1, w2, b2,
      out);
}